// AnalyticalTC_33414845563115
// MI455X (gfx1250) — compile-verified
//
#include <hip/hip_runtime.h>

typedef __attribute__((ext_vector_type(2))) float v2f;
typedef __attribute__((ext_vector_type(8))) float v8f;

#define BATCH 20000
#define CH 64
#define DD 27
#define WAVES_PER_BLOCK 8
#define WAVE_LDS 1760  // 1728 T1 floats + 27 T2 floats + pad (keeps 16B alignment per wave)

// out layout (concat of reference tuple):
//   out1: BATCH*CH*81  (meta1, (B,C,3,3,3,3))
//   out2: BATCH*CH*9   (meta2, (B,C,3,3))
//   out3: BATCH*CH*1   (meta3, (B,C))
__global__ __launch_bounds__(256) void tc33_kernel(const float* __restrict__ T1,
                                                   const float* __restrict__ T2,
                                                   float* __restrict__ out) {
    __shared__ float lds[WAVES_PER_BLOCK * WAVE_LDS];
    const int tid  = threadIdx.x;
    const int w    = tid >> 5;
    const int lane = tid & 31;
    const int b    = blockIdx.x * WAVES_PER_BLOCK + w;  // 2500*8 == 20000 exactly

    float* wl = &lds[w * WAVE_LDS];

    // ---- Stage T1[b] (1728 floats = 432 float4, 16B aligned since 6912B per b) ----
    {
        const float4* src = (const float4*)(T1 + (size_t)b * (CH * DD));
        float4*       dst = (float4*)wl;
        #pragma unroll
        for (int s = 0; s < 14; ++s) {
            int idx = s * 32 + lane;
            if (idx < 432) dst[idx] = src[idx];
        }
        if (lane < DD) wl[1728 + lane] = T2[(size_t)b * DD + lane];
    }
    __syncthreads();

    // T2 row into registers (constant indices only -> stays in VGPRs)
    float t2r[DD];
    #pragma unroll
    for (int d = 0; d < DD; ++d) t2r[d] = wl[1728 + d];

    float* out2 = out + (size_t)BATCH * CH * 81;
    float* out3 = out + (size_t)BATCH * CH * 90;

    const float n1 = 0.5773502691896258f;   // 1/sqrt(3)
    const float n2 = 1.0f / 3.0f;           // 1/sqrt(9)
    const float n3 = 0.19245008972987526f;  // 1/sqrt(27)

    // ---- meta3 (27-dot) and meta2 (3x9 @ 9x3), two channels per lane ----
    #pragma unroll
    for (int cc = 0; cc < 2; ++cc) {
        int c = lane + cc * 32;
        const float* row = wl + c * DD;
        float acc = 0.f;
        #pragma unroll
        for (int d = 0; d < DD; ++d) acc += row[d] * t2r[d];
        out3[(size_t)b * CH + c] = acc * n3;

        #pragma unroll
        for (int i = 0; i < 3; ++i) {
            #pragma unroll
            for (int j = 0; j < 3; ++j) {
                float a2 = 0.f;
                #pragma unroll
                for (int k = 0; k < 9; ++k) a2 += row[i * 9 + k] * t2r[k * 3 + j];
                out2[((size_t)b * CH + c) * 9 + i * 3 + j] = a2 * n2;
            }
        }
    }

    // ---- meta1: per-b GEMM (576 x 3) @ (3 x 9) via V_WMMA_F32_16X16X4_F32 ----
    // A row r = c*9+i maps to LDS float offset c*27+i*3 == 3*r == 48*t + 3*n,
    // so the per-lane LDS base is loop-invariant and tile offsets are immediates.
    const int  half = lane >> 4;   // 0: K={0,1} half / D rows 0-7;  1: K={2,3} half / D rows 8-15
    const int  n    = lane & 15;   // N column (valid if < 9)
    const bool act  = (n < 9);

    // B operand (4x16, K padded 3->4), branchless, loop-invariant:
    //   VGPR0 holds row K=2*half, VGPR1 holds row K=1 (half 0) / K=3==0 (half 1)
    v2f bmat;
    bmat.x = act              ? wl[1728 + 18 * half + n] : 0.f;
    bmat.y = (act && !half)   ? wl[1728 + 9 + n]         : 0.f;

    const float* abase = wl + 3 * n;                                  // + 48*t, +k
    float*       obase = out + (size_t)b * (CH * 81) + n + 72 * half; // + t*144 + v*9

    #pragma unroll
    for (int t = 0; t < 36; ++t) {
        // Unconditional loads (uniform exec), then lane-half select: no divergence.
        float a0 = abase[48 * t + 0];
        float a1 = abase[48 * t + 1];
        float a2 = abase[48 * t + 2];
        v2f amat;
        amat.x = half ? a2  : a0;
        amat.y = half ? 0.f : a1;

        v8f acc = {};
        acc = __builtin_amdgcn_wmma_f32_16x16x4_f32(false, amat, false, bmat,
                                                    (short)0, acc, false, false);

        // D: VGPR v, lanes<16 -> M=v, lanes>=16 -> M=v+8; out row = t*16+M, col = n
        if (act) {
            #pragma unroll
            for (int v = 0; v < 8; ++v) {
                obase[t * 144 + v * 9] = acc[v] * n1;
            }
        }
    }
}

extern "C" void kernel_launch(void* const* d_in, const int* in_sizes, int n_in,
                              void* d_out, int out_size, void* d_ws, size_t ws_size,
                              hipStream_t stream) {
    const float* T1 = (const float*)d_in[0];
    const float* T2 = (const float*)d_in[1];
    float* out = (float*)d_out;
    dim3 grid(BATCH / WAVES_PER_BLOCK);  // 2500 blocks
    dim3 block(256);                     // 8 wave32 per block, one b per wave
    hipLaunchKernelGGL(tc33_kernel, grid, block, 0, stream, T1, T2, out);
}